// BasicQuantumDQN_64579128262685
// MI455X (gfx1250) — compile-verified
//
#include <hip/hip_runtime.h>
#include <hip/hip_bf16.h>

// ---------------------------------------------------------------------------
// Quantum DQN circuit simulator for MI455X (gfx1250).
//   grid = 512 blocks (one per batch sample), 256 threads (8 wave32).
//   State vector (2^14 complex = 128KB as separate re/im f32) lives in LDS
//   for the whole circuit -> no HBM traffic for the state.
//   Each layer: 14 Rot gates fused into four 16x16 complex unitaries
//   (tensor products over qubit groups), applied with chained
//   V_WMMA_F32_16X16X4_F32 (f32 matrix path keeps complex64-level accuracy).
// ---------------------------------------------------------------------------

typedef __attribute__((ext_vector_type(2))) float v2f;
typedef __attribute__((ext_vector_type(8))) float v8f;

#define NQ      14
#define NSTATE  (1 << NQ)        // 16384
#define NLAYERS 6
#define NACT    8
#define NTHR    256

// LDS layout (floats):
//   sre[16384] | sim[16384] | rot[14*8] | Ure[256] | Uim[256]
//   | cs[14] | sn[14] | zE[14]
#define OFF_SRE  0
#define OFF_SIM  16384
#define OFF_ROT  32768
#define OFF_URE  32880
#define OFF_UIM  33136
#define OFF_CS   33392
#define OFF_SN   33406
#define OFF_ZE   33420
#define LDS_FLOATS 33434

__global__ __launch_bounds__(NTHR) void quantum_dqn_kernel(
    const float* __restrict__ x,        // [512,14]
    const float* __restrict__ w,        // [6,14,3]
    const float* __restrict__ Wh,       // [8,14]
    const float* __restrict__ bh,       // [8]
    float* __restrict__ out)            // [512,8]
{
    extern __shared__ float lds[];
    float* sre  = lds + OFF_SRE;
    float* sim  = lds + OFF_SIM;
    float* rotm = lds + OFF_ROT;   // [14][4 entries][re,im]
    float* Ure  = lds + OFF_URE;   // 16x16
    float* Uim  = lds + OFF_UIM;
    float* csA  = lds + OFF_CS;
    float* snA  = lds + OFF_SN;
    float* zE   = lds + OFF_ZE;

    const int tid  = threadIdx.x;
    const int lane = tid & 31;
    const int wave = tid >> 5;
    const int blk  = blockIdx.x;

    // ---- phase 0: per-sample angle-embedding cos/sin, zero Z accumulators
    if (tid < NQ) {
        float xi = x[blk * NQ + tid];
        csA[tid] = __cosf(0.5f * xi);
        snA[tid] = __sinf(0.5f * xi);
        zE[tid]  = 0.0f;
    }
    __syncthreads();

    // ---- phase 1: product-state init (RY embedding applied analytically)
    for (int k = tid; k < NSTATE; k += NTHR) {
        float amp = 1.0f;
        #pragma unroll
        for (int i = 0; i < NQ; ++i) {
            amp *= ((k >> (NQ - 1 - i)) & 1) ? snA[i] : csA[i];
        }
        sre[k] = amp;
        sim[k] = 0.0f;
    }
    __syncthreads();

    const int mrow  = lane & 15;          // M (or N) index inside a fragment
    const int khalf = (lane >> 4) * 2;    // K sub-offset per half-wave

    // ---- phase 2: entangling layers
    for (int l = 0; l < NLAYERS; ++l) {
        // 2a: 2x2 Rot matrices, one per qubit (PennyLane RZ*RY*RZ)
        if (tid < NQ) {
            const float* wl = w + (l * NQ + tid) * 3;
            float phi = wl[0], th = wl[1], om = wl[2];
            float c  = __cosf(0.5f * th), s  = __sinf(0.5f * th);
            float ap = -0.5f * (phi + om), am = -0.5f * (phi - om);
            float epr = __cosf(ap), epi = __sinf(ap);
            float emr = __cosf(am), emi = __sinf(am);
            float* R = &rotm[tid * 8];
            R[0] =  epr * c;  R[1] =  epi * c;   // R00 = ep*c
            R[2] = -emr * s;  R[3] =  emi * s;   // R01 = -conj(em)*s
            R[4] =  emr * s;  R[5] =  emi * s;   // R10 = em*s
            R[6] =  epr * c;  R[7] = -epi * c;   // R11 = conj(ep)*c
        }
        __syncthreads();

        // 2b: four fused 16x16 group unitaries (last group pads id on 10,11)
        const int g0s[4] = {0, 4, 8, 10};
        for (int g = 0; g < 4; ++g) {
            const int g0 = g0s[g];

            // build U = R_{g0} (x) R_{g0+1} (x) R_{g0+2} (x) R_{g0+3}
            {
                int ar = tid >> 4, bc = tid & 15;
                float ur = 1.0f, ui = 0.0f;
                #pragma unroll
                for (int t = 0; t < 4; ++t) {
                    int ra = (ar >> (3 - t)) & 1;
                    int rb = (bc >> (3 - t)) & 1;
                    float er, ei;
                    if (g == 3 && t < 2) {            // identity padding
                        er = (ra == rb) ? 1.0f : 0.0f; ei = 0.0f;
                    } else {
                        const float* R = &rotm[(g0 + t) * 8 + (ra * 2 + rb) * 2];
                        er = R[0]; ei = R[1];
                    }
                    float nr = ur * er - ui * ei;
                    ui = ur * ei + ui * er;
                    ur = nr;
                }
                Ure[tid] = ur;
                Uim[tid] = ui;
            }
            __syncthreads();

            // A-matrix fragments (ISA 32-bit 16x4 layout: lanes 0-15 K=0,1;
            // lanes 16-31 K=2,3), chained over 4 K-chunks for K=16.
            v2f aRe[4], aIm[4], aIn[4];
            #pragma unroll
            for (int kc = 0; kc < 4; ++kc) {
                int k0 = kc * 4 + khalf;
                v2f t0, t1;
                t0[0] = Ure[mrow * 16 + k0]; t0[1] = Ure[mrow * 16 + k0 + 1];
                t1[0] = Uim[mrow * 16 + k0]; t1[1] = Uim[mrow * 16 + k0 + 1];
                aRe[kc] = t0;
                aIm[kc] = t1;
                aIn[kc] = -t1;                 // -Im(U) for the Re accumulation
            }

            // apply U to state viewed as [2^g0, 16, 2^(10-g0)]
            const int shq = 10 - g0;
            const int str = 1 << shq;          // stride between group rows
            for (int tile = wave; tile < 64; tile += 8) {
                int c    = tile * 16 + mrow;   // this lane's column
                int p    = c >> shq;
                int q    = c & (str - 1);
                int base = (p << (14 - g0)) + q;

                v2f bRe[4], bIm[4];
                #pragma unroll
                for (int kc = 0; kc < 4; ++kc) {
                    int k0 = kc * 4 + khalf;
                    v2f t0, t1;
                    t0[0] = sre[base + k0 * str];
                    t0[1] = sre[base + (k0 + 1) * str];
                    t1[0] = sim[base + k0 * str];
                    t1[1] = sim[base + (k0 + 1) * str];
                    bRe[kc] = t0;
                    bIm[kc] = t1;
                }

                v8f cr = {}, ci = {};
                #pragma unroll
                for (int kc = 0; kc < 4; ++kc)      // Re += Ur*Sr
                    cr = __builtin_amdgcn_wmma_f32_16x16x4_f32(
                        false, aRe[kc], false, bRe[kc], (short)0, cr, false, false);
                #pragma unroll
                for (int kc = 0; kc < 4; ++kc)      // Re += (-Ui)*Si
                    cr = __builtin_amdgcn_wmma_f32_16x16x4_f32(
                        false, aIn[kc], false, bIm[kc], (short)0, cr, false, false);
                #pragma unroll
                for (int kc = 0; kc < 4; ++kc)      // Im += Ur*Si
                    ci = __builtin_amdgcn_wmma_f32_16x16x4_f32(
                        false, aRe[kc], false, bIm[kc], (short)0, ci, false, false);
                #pragma unroll
                for (int kc = 0; kc < 4; ++kc)      // Im += Ui*Sr
                    ci = __builtin_amdgcn_wmma_f32_16x16x4_f32(
                        false, aIm[kc], false, bRe[kc], (short)0, ci, false, false);

                // write back D (C/D layout: vgpr r -> rows r / r+8)
                #pragma unroll
                for (int r = 0; r < 8; ++r) {
                    int m = r + (lane >> 4) * 8;
                    int addr = base + m * str;
                    sre[addr] = cr[r];
                    sim[addr] = ci[r];
                }
            }
            __syncthreads();
        }

        // 2c: fused CZ chain — sign flip when adjacent wire bits are both 1
        for (int k = tid; k < NSTATE; k += NTHR) {
            if (__popc(k & (k >> 1)) & 1) {
                sre[k] = -sre[k];
                sim[k] = -sim[k];
            }
        }
        __syncthreads();
    }

    // ---- phase 3: <Z_i> = sum_k |amp|^2 * (bit_i ? -1 : +1)
    {
        float zl[NQ];
        #pragma unroll
        for (int i = 0; i < NQ; ++i) zl[i] = 0.0f;
        for (int k = tid; k < NSTATE; k += NTHR) {
            float pr = sre[k] * sre[k] + sim[k] * sim[k];
            #pragma unroll
            for (int i = 0; i < NQ; ++i)
                zl[i] += ((k >> (NQ - 1 - i)) & 1) ? -pr : pr;
        }
        #pragma unroll
        for (int i = 0; i < NQ; ++i) {
            #pragma unroll
            for (int off = 16; off > 0; off >>= 1)
                zl[i] += __shfl_down(zl[i], off, 32);
        }
        if (lane == 0) {
            #pragma unroll
            for (int i = 0; i < NQ; ++i)
                atomicAdd(&zE[i], zl[i]);
        }
    }
    __syncthreads();

    // ---- phase 4: linear head out = z @ W^T + b
    if (tid < NACT) {
        float acc = bh[tid];
        #pragma unroll
        for (int i = 0; i < NQ; ++i)
            acc += zE[i] * Wh[tid * NQ + i];
        out[blk * NACT + tid] = acc;
    }
}

extern "C" void kernel_launch(void* const* d_in, const int* in_sizes, int n_in,
                              void* d_out, int out_size, void* d_ws, size_t ws_size,
                              hipStream_t stream) {
    (void)in_sizes; (void)n_in; (void)d_ws; (void)ws_size; (void)out_size;
    const float* x  = (const float*)d_in[0];   // [512,14]
    const float* w  = (const float*)d_in[1];   // [6,14,3]
    const float* Wh = (const float*)d_in[2];   // [8,14]
    const float* bh = (const float*)d_in[3];   // [8]
    float* out = (float*)d_out;                // [512,8]

    const size_t ldsBytes = (size_t)LDS_FLOATS * sizeof(float);  // ~131 KB
    quantum_dqn_kernel<<<512, NTHR, ldsBytes, stream>>>(x, w, Wh, bh, out);
}